// GridEncoder_44212393345206
// MI455X (gfx1250) — compile-verified
//
#include <hip/hip_runtime.h>
#include <hip/hip_bf16.h>

// ---------------------------------------------------------------------------
// Instant-NGP grid encode, specialized for the reference's fixed config:
//   D=3, L=16, C=2, base_res=16, scale=2.0, log2_hash=19, align_corners=False
// Per-level constants are compile-time:
//   scale_l   = 16*2^l - 1          (exact in f32)
//   res_l     = 16*2^l              (dense stride)
//   levels 0..2 dense (no mod needed: max dense index < table size)
//   levels 3..15 hashed, hsize = 2^19  ->  mod == & 0x7FFFF
// ---------------------------------------------------------------------------

#define PRIME1 2654435761u
#define PRIME2 805459861u
#define HASH_MASK 0x7FFFFu
#define L0_ENTRIES 4920          // level-0 table entries (float2 each) = 38.4 KB

static __device__ constexpr unsigned LEVEL_OFF[16] = {
    0u,       4920u,    40864u,   315496u,  839784u,  1364072u,
    1888360u, 2412648u, 2936936u, 3461224u, 3985512u, 4509800u,
    5034088u, 5558376u, 6082664u, 6606952u};

template <int LVL>
__device__ __forceinline__ void encode_point(float x, float y, float z,
                                             const float2* __restrict__ tab,
                                             float2* __restrict__ out2, int b) {
    constexpr unsigned RES = 16u << LVL;
    constexpr float SCALE = (float)(RES - 1u);

    // pos = x*scale + 0.5 with explicit mul-then-add (match XLA rounding so
    // floor() picks the same cell as the reference)
    float px = __fadd_rn(__fmul_rn(x, SCALE), 0.5f);
    float py = __fadd_rn(__fmul_rn(y, SCALE), 0.5f);
    float pz = __fadd_rn(__fmul_rn(z, SCALE), 0.5f);
    float gx = floorf(px), gy = floorf(py), gz = floorf(pz);
    float fx = __fsub_rn(px, gx), fy = __fsub_rn(py, gy), fz = __fsub_rn(pz, gz);
    unsigned ix = (unsigned)(int)gx, iy = (unsigned)(int)gy, iz = (unsigned)(int)gz;

    unsigned idx[8];
    if constexpr (LVL >= 3) {
        // fast_hash with prime[0]==1; share the multiplies across corners
        unsigned ax0 = ix, ax1 = ix + 1u;
        unsigned by0 = iy * PRIME1, by1 = by0 + PRIME1;
        unsigned cz0 = iz * PRIME2, cz1 = cz0 + PRIME2;
#pragma unroll
        for (int c = 0; c < 8; ++c) {
            unsigned h = ((c & 1) ? ax1 : ax0) ^ ((c & 2) ? by1 : by0) ^
                         ((c & 4) ? cz1 : cz0);
            idx[c] = h & HASH_MASK;
        }
    } else {
        // dense: c0 + c1*R + c2*R^2 ; proven < table size, so no mod
#pragma unroll
        for (int c = 0; c < 8; ++c) {
            idx[c] = (ix + (unsigned)(c & 1)) +
                     (iy + (unsigned)((c >> 1) & 1)) * RES +
                     (iz + (unsigned)((c >> 2) & 1)) * (RES * RES);
        }
    }

    // issue all 8 independent b64 gathers before consuming (MLP)
    float2 v[8];
#pragma unroll
    for (int c = 0; c < 8; ++c) v[c] = tab[idx[c]];

    float wx[2] = {__fsub_rn(1.0f, fx), fx};
    float wy[2] = {__fsub_rn(1.0f, fy), fy};
    float wz[2] = {__fsub_rn(1.0f, fz), fz};
    float a0 = 0.0f, a1 = 0.0f;
#pragma unroll
    for (int c = 0; c < 8; ++c) {
        float w = __fmul_rn(__fmul_rn(wx[c & 1], wy[(c >> 1) & 1]), wz[(c >> 2) & 1]);
        a0 = fmaf(w, v[c].x, a0);
        a1 = fmaf(w, v[c].y, a1);
    }
    out2[(size_t)b * 16 + LVL] = make_float2(a0, a1);
}

__global__ __launch_bounds__(256) void GridEncoder_44212393345206_kernel(
    const float* __restrict__ xyz, const float2* __restrict__ emb,
    float2* __restrict__ out2, int nB) {
    __shared__ float2 s_l0[L0_ENTRIES];   // 38.4 KB of the 320 KB WGP LDS
    const int lvl = blockIdx.y;

    if (lvl == 0) {
        // Bulk-stage the whole level-0 table into LDS with the CDNA5 async
        // global->LDS path (tracked by ASYNCcnt, per-lane b128 transfers).
        unsigned lds_base = (unsigned)(unsigned long long)(void*)&s_l0[0];
        unsigned long long gbase = (unsigned long long)(const void*)emb; // L0 offset == 0
        for (int i = threadIdx.x; i < (L0_ENTRIES * 8) / 16; i += blockDim.x) {
            unsigned la = lds_base + (unsigned)i * 16u;
            unsigned long long ga = gbase + (unsigned long long)i * 16ull;
            asm volatile("global_load_async_to_lds_b128 %0, %1, off"
                         :: "v"(la), "v"(ga) : "memory");
        }
        asm volatile("s_wait_asynccnt 0" ::: "memory");
        __syncthreads();
    }

    const int stride = gridDim.x * blockDim.x;
    const int b0 = blockIdx.x * blockDim.x + threadIdx.x;

#define LEVEL_CASE(L)                                                         \
    case L: {                                                                 \
        const float2* __restrict__ tab =                                      \
            (L == 0) ? (const float2*)s_l0 : (emb + LEVEL_OFF[L]);            \
        for (int b = b0; b < nB; b += stride) {                               \
            float x = xyz[3 * b + 0];                                         \
            float y = xyz[3 * b + 1];                                         \
            float z = xyz[3 * b + 2];                                         \
            encode_point<L>(x, y, z, tab, out2, b);                           \
        }                                                                     \
    } break;

    switch (lvl) {
        LEVEL_CASE(0)  LEVEL_CASE(1)  LEVEL_CASE(2)  LEVEL_CASE(3)
        LEVEL_CASE(4)  LEVEL_CASE(5)  LEVEL_CASE(6)  LEVEL_CASE(7)
        LEVEL_CASE(8)  LEVEL_CASE(9)  LEVEL_CASE(10) LEVEL_CASE(11)
        LEVEL_CASE(12) LEVEL_CASE(13) LEVEL_CASE(14) LEVEL_CASE(15)
        default: break;
    }
#undef LEVEL_CASE
}

extern "C" void kernel_launch(void* const* d_in, const int* in_sizes, int n_in,
                              void* d_out, int out_size, void* d_ws, size_t ws_size,
                              hipStream_t stream) {
    (void)n_in; (void)d_ws; (void)ws_size; (void)out_size;
    const float*  xyz = (const float*)d_in[0];     // [B,3] f32
    const float2* emb = (const float2*)d_in[1];    // [offsets[-1],2] f32
    // d_in[2] (offsets) is deterministic from the reference config; hardcoded.
    float2* out2 = (float2*)d_out;                 // [B,16] float2 == [B,32] f32
    const int nB = in_sizes[0] / 3;

    // grid.y = level (uniform per block, fully specialized); grid.x blocks
    // grid-stride over points so level-0 blocks amortize their LDS staging.
    dim3 grid(256, 16, 1);
    dim3 block(256, 1, 1);
    GridEncoder_44212393345206_kernel<<<grid, block, 0, stream>>>(xyz, emb, out2, nB);
}